// SpatiallySparseByChannel_50173807952757
// MI455X (gfx1250) — compile-verified
//
#include <hip/hip_runtime.h>
#include <stdint.h>

// ---------------- problem constants (from reference) ----------------
#define N_ 32
#define C_ 256
#define L_ 8192
#define NL_ (N_ * L_)            // 262144 values per channel
#define K_SEL 235929u            // max(1, int(NL * 0.9)) ; 1-indexed kth smallest
#define MOMENTUM_ 0.1f

#define TPB 256
#define HIST_BINS 65536
#define STAGE_ELEMS 4096         // floats per stage buffer (16 KB)
#define CHUNKS (NL_ / STAGE_ELEMS) // 64 chunks of 4096 floats per channel
// dynamic LDS: 256KB hist + 2x16KB stage + scan scratch  (< 320KB WGP LDS)
#define SMEM_BYTES (HIST_BINS * 4 + 2 * STAGE_ELEMS * 4 + 264 * 4)

// ---------------- gfx1250 async global->LDS helpers ----------------
typedef int vi4 __attribute__((vector_size(16)));
typedef __attribute__((address_space(1))) vi4* as1_vi4_ptr;   // global
typedef __attribute__((address_space(3))) vi4* as3_vi4_ptr;   // LDS

__device__ __forceinline__ void async_copy_b128(const float* gsrc, float* lds_dst) {
#if __has_builtin(__builtin_amdgcn_global_load_async_to_lds_b128)
    __builtin_amdgcn_global_load_async_to_lds_b128(
        (as1_vi4_ptr)(uintptr_t)gsrc,
        (as3_vi4_ptr)(uint32_t)(uintptr_t)lds_dst,   // low 32 bits == LDS offset
        0, 0);
#else
    // GV mode: VDST = LDS byte offset, VADDR = 64-bit global address, saddr = off
    uint32_t lds_off = (uint32_t)(uintptr_t)lds_dst;
    asm volatile("global_load_async_to_lds_b128 %0, %1, off"
                 :: "v"(lds_off), "v"(gsrc) : "memory");
#endif
}

template <int NCNT>
__device__ __forceinline__ void wait_asynccnt() {
#if __has_builtin(__builtin_amdgcn_s_wait_asynccnt)
    __builtin_amdgcn_s_wait_asynccnt(NCNT);
#else
    asm volatile("s_wait_asynccnt %0" :: "i"(NCNT) : "memory");
#endif
    asm volatile("" ::: "memory"); // keep LDS reads below the wait
}

// ---------------- order-preserving float <-> uint mapping ----------------
__device__ __forceinline__ uint32_t sortable_u32(float f) {
    uint32_t u = __float_as_uint(f);
    return u ^ ((uint32_t)((int32_t)u >> 31) | 0x80000000u);
}
__device__ __forceinline__ float unsortable_f32(uint32_t u) {
    return (u & 0x80000000u) ? __uint_as_float(u ^ 0x80000000u)
                             : __uint_as_float(~u);
}

// Issue one 4096-float chunk of channel c as 4 async b128 loads per thread.
__device__ __forceinline__ void issue_chunk(const float* __restrict__ x, int c,
                                            int chunk, float* stage_buf, int t) {
    const size_t row   = (size_t)(chunk >> 1);           // 2 chunks per L-row
    const size_t half  = (size_t)(chunk & 1) * STAGE_ELEMS;
    const float* base  = x + row * (size_t)C_ * L_ + (size_t)c * L_ + half;
#pragma unroll
    for (int j = 0; j < 4; ++j) {
        const int e = (t + j * TPB) * 4;                 // float offset of b128
        async_copy_b128(base + e, stage_buf + e);
    }
}

// Cooperative: find bin b (0..65535) s.t. cum(hist[0..b-1]) < K <= cum(hist[0..b]).
// Returns bin; *remK = K - cum(hist[0..b-1])  (1-indexed rank inside the bin).
__device__ uint32_t scan_select(uint32_t* hist, uint32_t* scanbuf,
                                uint32_t K, uint32_t* remK, int t) {
    uint32_t s = 0;
    for (int b = 0; b < 256; ++b) s += hist[t * 256 + b];
    scanbuf[t] = s;
    __syncthreads();
    if (t == 0) {
        uint32_t cum = 0, segbase = 0; int seg = -1;
        for (int i = 0; i < 256; ++i) {
            uint32_t v = scanbuf[i];
            if (seg < 0 && cum + v >= K) { seg = i; segbase = cum; }
            cum += v;
        }
        if (seg < 0) { seg = 255; segbase = cum - scanbuf[255]; }
        scanbuf[256] = (uint32_t)seg;
        scanbuf[257] = segbase;
    }
    __syncthreads();
    const int      seg     = (int)scanbuf[256];
    const uint32_t segbase = scanbuf[257];
    if (t == seg) {
        uint32_t cum = segbase; int bin = 255;
        for (int b = 0; b < 256; ++b) {
            uint32_t v = hist[seg * 256 + b];
            if (cum + v >= K) { bin = b; break; }
            cum += v;
        }
        scanbuf[258] = (uint32_t)(seg * 256 + bin);
        scanbuf[259] = K - cum;
    }
    __syncthreads();
    const uint32_t sel = scanbuf[258];
    *remK = scanbuf[259];
    __syncthreads();   // scanbuf reusable by caller
    return sel;
}

// ---------------- kernel 1: exact per-channel kth value + EMA ----------------
extern "C" __global__ __launch_bounds__(TPB)
void select_kth_kernel(const float* __restrict__ x,
                       const float* __restrict__ th_old,
                       float* __restrict__ th_new) {
    extern __shared__ unsigned char smem_raw[];
    uint32_t* hist    = (uint32_t*)smem_raw;                           // 65536
    float*    stage   = (float*)(smem_raw + HIST_BINS * 4);            // 2*4096
    uint32_t* scanbuf = (uint32_t*)(smem_raw + HIST_BINS * 4 + 2 * STAGE_ELEMS * 4);

    const int c = blockIdx.x;
    const int t = threadIdx.x;

    for (int i = t; i < HIST_BINS; i += TPB) hist[i] = 0u;
    __syncthreads();

    // ---- pass 1: hi-16 histogram via double-buffered async global->LDS ----
    issue_chunk(x, c, 0, stage, t);
    for (int k = 0; k < CHUNKS; ++k) {
        const int buf = k & 1;
        if (k + 1 < CHUNKS) {
            issue_chunk(x, c, k + 1, stage + ((k + 1) & 1) * STAGE_ELEMS, t);
            wait_asynccnt<4>();   // chunk k's 4 ops done (loads retire in order)
        } else {
            wait_asynccnt<0>();
        }
        const float4* sb = (const float4*)(stage + buf * STAGE_ELEMS);
#pragma unroll
        for (int j = 0; j < 4; ++j) {
            float4 v = sb[t + j * TPB];   // exactly the lanes this thread staged
            atomicAdd(&hist[sortable_u32(v.x) >> 16], 1u);
            atomicAdd(&hist[sortable_u32(v.y) >> 16], 1u);
            atomicAdd(&hist[sortable_u32(v.z) >> 16], 1u);
            atomicAdd(&hist[sortable_u32(v.w) >> 16], 1u);
        }
    }
    __syncthreads();

    uint32_t rem;
    const uint32_t hiSel = scan_select(hist, scanbuf, K_SEL, &rem, t);

    // ---- pass 2: lo-16 histogram restricted to the selected hi bucket ----
    for (int i = t; i < HIST_BINS; i += TPB) hist[i] = 0u;
    __syncthreads();

    for (int n = 0; n < N_; ++n) {
        const float4* row =
            (const float4*)(x + (size_t)n * C_ * L_ + (size_t)c * L_);
        for (int i = t; i < L_ / 4; i += TPB) {
            float4 v = row[i];
            uint32_t u;
            u = sortable_u32(v.x); if ((u >> 16) == hiSel) atomicAdd(&hist[u & 0xFFFFu], 1u);
            u = sortable_u32(v.y); if ((u >> 16) == hiSel) atomicAdd(&hist[u & 0xFFFFu], 1u);
            u = sortable_u32(v.z); if ((u >> 16) == hiSel) atomicAdd(&hist[u & 0xFFFFu], 1u);
            u = sortable_u32(v.w); if ((u >> 16) == hiSel) atomicAdd(&hist[u & 0xFFFFu], 1u);
        }
    }
    __syncthreads();

    uint32_t rem2;
    const uint32_t loSel = scan_select(hist, scanbuf, rem, &rem2, t);

    if (t == 0) {
        const uint32_t ku  = (hiSel << 16) | (loSel & 0xFFFFu);
        const float    kth = unsortable_f32(ku);
        th_new[c] = th_old[c] * (1.0f - MOMENTUM_) + kth * MOMENTUM_;
    }
}

// ---------------- kernel 2: out = relu(x - th[c]), float4 streaming ----------------
extern "C" __global__ __launch_bounds__(TPB)
void fused_sub_relu_kernel(const float4* __restrict__ x,
                           const float* __restrict__ th,
                           float4* __restrict__ out, int total4) {
    const int stride = gridDim.x * blockDim.x;
    for (int i = blockIdx.x * blockDim.x + threadIdx.x; i < total4; i += stride) {
        const int cidx = (i >> 11) & (C_ - 1);   // L/4 = 2048 float4 per channel row
        const float tv = th[cidx];
        float4 v = x[i];
        float4 r;
        r.x = fmaxf(v.x - tv, 0.0f);
        r.y = fmaxf(v.y - tv, 0.0f);
        r.z = fmaxf(v.z - tv, 0.0f);
        r.w = fmaxf(v.w - tv, 0.0f);
        out[i] = r;
    }
}

extern "C" void kernel_launch(void* const* d_in, const int* in_sizes, int n_in,
                              void* d_out, int out_size, void* d_ws, size_t ws_size,
                              hipStream_t stream) {
    const float* x      = (const float*)d_in[0];   // [N, C, L] fp32
    const float* th_old = (const float*)d_in[1];   // [C] fp32
    float*       out    = (float*)d_out;           // [N, C, L] fp32
    float*       th_new = (float*)d_ws;            // [C] scratch

    // one workgroup per channel; ~295 KB dynamic LDS per WG (CDNA5: 320 KB/WGP)
    select_kth_kernel<<<C_, TPB, SMEM_BYTES, stream>>>(x, th_old, th_new);

    const int total4 = (N_ * C_ * L_) / 4;  // 16,777,216 float4
    fused_sub_relu_kernel<<<8192, TPB, 0, stream>>>(
        (const float4*)x, th_new, (float4*)out, total4);
}